// BatchedProcrustes_23158463660117
// MI455X (gfx1250) — compile-verified
//
#include <hip/hip_runtime.h>
#include <hip/hip_bf16.h>

// ---------------------------------------------------------------------------
// Batched Procrustes (Kabsch) for MI455X (gfx1250, wave32).
//   d_in[0] = src  [N,3] f32
//   d_in[1] = tgt  [N,3] f32
//   d_in[2] = bidx [N]   i32 (sorted)
//   d_out   = aligned [N,3] f32  ++  R [B,3,3] f32  ++  t [B,3] f32
//   d_ws    = moments [B,16] f32 (count, sum_s[3], sum_t[3], cross[9])
// ---------------------------------------------------------------------------

typedef __attribute__((ext_vector_type(2))) float v2f;
typedef __attribute__((ext_vector_type(8))) float v8f;

// ------------------------------ zero workspace ------------------------------
__global__ void bp_zero_kernel(float* __restrict__ p, int n) {
    int i = blockIdx.x * blockDim.x + threadIdx.x;
    if (i < n) p[i] = 0.0f;
}

// ------------------------- phase 1: segmented moments -----------------------
// Wave-contiguous chunks, lane-interleaved access (coalesced), register-local
// accumulation exploiting sorted bidx; atomic flush only on segment change.
__device__ inline void bp_flush(float* __restrict__ mom, int b, const float* acc) {
    if (b < 0) return;
    float* p = mom + (size_t)b * 16;
#pragma unroll
    for (int m = 0; m < 16; ++m) atomicAdd(p + m, acc[m]);
}

__global__ void bp_moments_kernel(const float* __restrict__ src,
                                  const float* __restrict__ tgt,
                                  const int* __restrict__ bidx,
                                  float* __restrict__ mom, int N) {
    int tid      = blockIdx.x * blockDim.x + threadIdx.x;
    int nthreads = gridDim.x * blockDim.x;
    int wave = tid >> 5, lane = tid & 31;
    int nwaves = nthreads >> 5;
    long long ppw   = ((long long)N + nwaves - 1) / nwaves;   // points per wave
    long long start = (long long)wave * ppw;
    long long end   = start + ppw;
    if (end > N) end = N;

    float acc[16];
#pragma unroll
    for (int m = 0; m < 16; ++m) acc[m] = 0.0f;
    int cur = -1;

    for (long long i = start + lane; i < end; i += 32) {
        // stream prefetch ~1.5KB ahead (global_prefetch_b8)
        __builtin_prefetch(src + 3 * (i + 128), 0, 0);
        __builtin_prefetch(tgt + 3 * (i + 128), 0, 0);

        int b = bidx[i];
        float sx = src[3 * i + 0], sy = src[3 * i + 1], sz = src[3 * i + 2];
        float tx = tgt[3 * i + 0], ty = tgt[3 * i + 1], tz = tgt[3 * i + 2];

        if (b != cur) {                 // sorted bidx => rare (segment boundary)
            bp_flush(mom, cur, acc);
#pragma unroll
            for (int m = 0; m < 16; ++m) acc[m] = 0.0f;
            cur = b;
        }
        acc[0] += 1.0f;
        acc[1] += sx;  acc[2] += sy;  acc[3] += sz;
        acc[4] += tx;  acc[5] += ty;  acc[6] += tz;
        acc[7]  = fmaf(sx, tx, acc[7]);   acc[8]  = fmaf(sx, ty, acc[8]);   acc[9]  = fmaf(sx, tz, acc[9]);
        acc[10] = fmaf(sy, tx, acc[10]);  acc[11] = fmaf(sy, ty, acc[11]);  acc[12] = fmaf(sy, tz, acc[12]);
        acc[13] = fmaf(sz, tx, acc[13]);  acc[14] = fmaf(sz, ty, acc[14]);  acc[15] = fmaf(sz, tz, acc[15]);
    }
    bp_flush(mom, cur, acc);
}

// ----------------------- phase 2: per-segment Kabsch ------------------------
// One thread per segment. Jacobi eigendecomposition of H^T H -> V, sigma;
// U = H V / sigma (re-orthogonalized); R = V diag(1,1,d) U^T; t = mu_t - R mu_s.
__global__ void bp_kabsch_kernel(const float* __restrict__ mom,
                                 float* __restrict__ outR,
                                 float* __restrict__ outT, int B) {
    int b = blockIdx.x * blockDim.x + threadIdx.x;
    if (b >= B) return;
    const float* m = mom + (size_t)b * 16;

    float cnt = m[0];
    float n   = fmaxf(cnt, 1.0f);
    float mus[3] = { m[1] / n, m[2] / n, m[3] / n };
    float mut[3] = { m[4] / n, m[5] / n, m[6] / n };

    float H[3][3];
#pragma unroll
    for (int j = 0; j < 3; ++j)
#pragma unroll
        for (int k = 0; k < 3; ++k)
            H[j][k] = m[7 + 3 * j + k] - cnt * mus[j] * mut[k];

    // A = H^T H (symmetric PSD)
    float A[3][3];
#pragma unroll
    for (int j = 0; j < 3; ++j)
#pragma unroll
        for (int k = 0; k < 3; ++k)
            A[j][k] = H[0][j] * H[0][k] + H[1][j] * H[1][k] + H[2][j] * H[2][k];

    float V[3][3] = { {1,0,0},{0,1,0},{0,0,1} };
    const int JP[3] = {0, 0, 1}, JQ[3] = {1, 2, 2};
#pragma unroll
    for (int sweep = 0; sweep < 10; ++sweep) {
#pragma unroll
        for (int r = 0; r < 3; ++r) {
            int p = JP[r], q = JQ[r];
            float apq = A[p][q];
            if (fabsf(apq) > 1e-12f) {
                float theta = (A[q][q] - A[p][p]) / (2.0f * apq);
                float tr = copysignf(1.0f, theta) / (fabsf(theta) + sqrtf(theta * theta + 1.0f));
                float c = 1.0f / sqrtf(tr * tr + 1.0f);
                float s = tr * c;
#pragma unroll
                for (int i = 0; i < 3; ++i) { float ap = A[i][p], aq = A[i][q];
                    A[i][p] = c * ap - s * aq;  A[i][q] = s * ap + c * aq; }
#pragma unroll
                for (int i = 0; i < 3; ++i) { float ap = A[p][i], aq = A[q][i];
                    A[p][i] = c * ap - s * aq;  A[q][i] = s * ap + c * aq; }
#pragma unroll
                for (int i = 0; i < 3; ++i) { float vp = V[i][p], vq = V[i][q];
                    V[i][p] = c * vp - s * vq;  V[i][q] = s * vp + c * vq; }
            }
        }
    }

    float lam[3] = { A[0][0], A[1][1], A[2][2] };
    // sort eigenpairs descending (columns of V follow)
#pragma unroll
    for (int a = 0; a < 2; ++a)
#pragma unroll
        for (int bb2 = 0; bb2 < 2 - a; ++bb2) {
            if (lam[bb2] < lam[bb2 + 1]) {
                float tl = lam[bb2]; lam[bb2] = lam[bb2 + 1]; lam[bb2 + 1] = tl;
#pragma unroll
                for (int i = 0; i < 3; ++i) { float tv = V[i][bb2]; V[i][bb2] = V[i][bb2 + 1]; V[i][bb2 + 1] = tv; }
            }
        }

    float sig0 = sqrtf(fmaxf(lam[0], 0.0f));
    float sig1 = sqrtf(fmaxf(lam[1], 0.0f));

    float U[3][3];
    // u0 = H v0 / sig0
    {
        float hv[3];
#pragma unroll
        for (int i = 0; i < 3; ++i)
            hv[i] = H[i][0] * V[0][0] + H[i][1] * V[1][0] + H[i][2] * V[2][0];
        float inv = (sig0 > 1e-8f) ? (1.0f / sig0) : 0.0f;
        U[0][0] = (sig0 > 1e-8f) ? hv[0] * inv : 1.0f;
        U[1][0] = hv[1] * inv;  U[2][0] = hv[2] * inv;
    }
    // u1 = H v1 / sig1, Gram-Schmidt vs u0
    {
        float hv[3];
#pragma unroll
        for (int i = 0; i < 3; ++i)
            hv[i] = H[i][0] * V[0][1] + H[i][1] * V[1][1] + H[i][2] * V[2][1];
        float inv = (sig1 > 1e-8f) ? (1.0f / sig1) : 0.0f;
        float u1[3] = { (sig1 > 1e-8f) ? hv[0] * inv : 0.0f,
                        (sig1 > 1e-8f) ? hv[1] * inv : 1.0f,
                        hv[2] * inv };
        float d01 = u1[0] * U[0][0] + u1[1] * U[1][0] + u1[2] * U[2][0];
        u1[0] -= d01 * U[0][0]; u1[1] -= d01 * U[1][0]; u1[2] -= d01 * U[2][0];
        float nn = sqrtf(u1[0]*u1[0] + u1[1]*u1[1] + u1[2]*u1[2]);
        float in2 = (nn > 1e-8f) ? 1.0f / nn : 0.0f;
        U[0][1] = u1[0] * in2; U[1][1] = u1[1] * in2; U[2][1] = u1[2] * in2;
        if (nn <= 1e-8f) { U[0][1] = -U[1][0]; U[1][1] = U[0][0]; U[2][1] = 0.0f; }
    }
    // u2 = u0 x u1 (det(U) = +1 by construction)
    U[0][2] = U[1][0] * U[2][1] - U[2][0] * U[1][1];
    U[1][2] = U[2][0] * U[0][1] - U[0][0] * U[2][1];
    U[2][2] = U[0][0] * U[1][1] - U[1][0] * U[0][1];

    float detV = V[0][0] * (V[1][1] * V[2][2] - V[2][1] * V[1][2])
               - V[0][1] * (V[1][0] * V[2][2] - V[2][0] * V[1][2])
               + V[0][2] * (V[1][0] * V[2][1] - V[2][0] * V[1][1]);
    float d = (detV >= 0.0f) ? 1.0f : -1.0f;   // det(U)=+1 -> sign(det(V U^T)) = sign(det V)

    float R[3][3], t[3];
#pragma unroll
    for (int i = 0; i < 3; ++i)
#pragma unroll
        for (int j = 0; j < 3; ++j)
            R[i][j] = V[i][0] * U[j][0] + V[i][1] * U[j][1] + d * V[i][2] * U[j][2];
#pragma unroll
    for (int i = 0; i < 3; ++i)
        t[i] = mut[i] - (R[i][0] * mus[0] + R[i][1] * mus[1] + R[i][2] * mus[2]);

    float* rp = outR + (size_t)b * 9;
#pragma unroll
    for (int i = 0; i < 3; ++i)
#pragma unroll
        for (int j = 0; j < 3; ++j) rp[3 * i + j] = R[i][j];
    float* tp = outT + (size_t)b * 3;
    tp[0] = t[0]; tp[1] = t[1]; tp[2] = t[2];
}

// --------------------- phase 3: apply (WMMA f32 16x16x4) --------------------
// Per 16-point tile: aligned_tile(4x16) = [R|t; 0 1](4x4) * [s;1](4x16).
// A layout (ISA 7.12.2, 32-bit 16x4): lane L<16 row M=L {VGPR0:K0, VGPR1:K1};
//                                     lane L>=16 row M=L-16 {VGPR0:K2, VGPR1:K3}.
// B layout (4x16, row per half-VGPR): VGPR0 {lanes0-15:K0 row, lanes16-31:K2 row},
//                                     VGPR1 {lanes0-15:K1 row, lanes16-31:K3 row}.
// C layout: VGPRm lanes0-15 = row M=m  (m=0..2 are x,y,z of the 16 points).
// A-fragment is cached in registers across tiles of the same segment
// (~128 consecutive tiles per segment since bidx is sorted).
__global__ void bp_apply_kernel(const float* __restrict__ src,
                                const int* __restrict__ bidx,
                                const float* __restrict__ Rg,
                                const float* __restrict__ Tg,
                                float* __restrict__ out, int N) {
    int tid      = blockIdx.x * blockDim.x + threadIdx.x;
    int nthreads = gridDim.x * blockDim.x;
    int wave = tid >> 5, lane = tid & 31;
    int nwaves = nthreads >> 5;
    int l15 = lane & 15;
    bool hi = lane >= 16;

    int ntiles = (N + 15) >> 4;
    int tpw = (ntiles + nwaves - 1) / nwaves;
    int tbeg = wave * tpw;                       // wave-uniform contiguous run
    int tend = tbeg + tpw; if (tend > ntiles) tend = ntiles;

    int   curb = -1;                             // cached segment of A-fragment
    float ax = 0.0f, ay = 0.0f;                  // cached A-fragment (this lane)

#pragma unroll 2
    for (int tile = tbeg; tile < tend; ++tile) { // wave-uniform trip count
        int p = tile << 4;
        int plast = p + 15 < N ? p + 15 : N - 1;
        int b0 = bidx[p];                        // sorted index: uniform iff ends match
        int b1 = bidx[plast];

        if (b0 == b1 && p + 15 < N) {
            // ---- WMMA fast path (covers ~99% of tiles) ----
            if (b0 != curb) {                    // wave-uniform: refresh A-fragment
                curb = b0;
                ax = 0.0f; ay = 0.0f;
                if (l15 < 3) {
                    const float* Rrow = Rg + (size_t)b0 * 9 + (size_t)l15 * 3;
                    if (!hi) { ax = Rrow[0]; ay = Rrow[1]; }                  // K=0,1
                    else     { ax = Rrow[2]; ay = Tg[(size_t)b0 * 3 + l15]; } // K=2,3(=t)
                }
            }
            int pi = p + l15;
            const float* sp = src + (size_t)3 * pi;
            float bx, by;
            if (!hi) { bx = sp[0]; by = sp[1]; }    // rows K=0 (x), K=1 (y)
            else     { bx = sp[2]; by = 1.0f;  }    // rows K=2 (z), K=3 (homog. 1)

            v2f a;  a.x  = ax; a.y  = ay;
            v2f bb; bb.x = bx; bb.y = by;
            v8f c = {};
            c = __builtin_amdgcn_wmma_f32_16x16x4_f32(
                    /*neg_a=*/false, a, /*neg_b=*/false, bb,
                    /*c_mod=*/(short)0, c, /*reuse_a=*/false, /*reuse_b=*/false);

            if (!hi) {
                float* op = out + (size_t)3 * pi;
                op[0] = c[0]; op[1] = c[1]; op[2] = c[2];  // rows M=0,1,2
            }
        } else {
            // ---- scalar fallback: segment-boundary or ragged tail tile ----
            int pi = p + l15;
            if (!hi && pi < N) {
                int b = bidx[pi];
                const float* Rb = Rg + (size_t)b * 9;
                const float* tb = Tg + (size_t)b * 3;
                const float* sp = src + (size_t)3 * pi;
                float sx = sp[0], sy = sp[1], sz = sp[2];
                float* op = out + (size_t)3 * pi;
                op[0] = fmaf(Rb[0], sx, fmaf(Rb[1], sy, fmaf(Rb[2], sz, tb[0])));
                op[1] = fmaf(Rb[3], sx, fmaf(Rb[4], sy, fmaf(Rb[5], sz, tb[1])));
                op[2] = fmaf(Rb[6], sx, fmaf(Rb[7], sy, fmaf(Rb[8], sz, tb[2])));
            }
        }
    }
}

// --------------------------------- launcher ---------------------------------
extern "C" void kernel_launch(void* const* d_in, const int* in_sizes, int n_in,
                              void* d_out, int out_size, void* d_ws, size_t ws_size,
                              hipStream_t stream) {
    const float* src  = (const float*)d_in[0];
    const float* tgt  = (const float*)d_in[1];
    const int*   bidx = (const int*)d_in[2];

    int N = in_sizes[0] / 3;
    int B = (out_size - 3 * N) / 12;           // out = N*3 + B*9 + B*3

    float* out_aligned = (float*)d_out;
    float* outR = out_aligned + (size_t)3 * N;
    float* outT = outR + (size_t)9 * B;
    float* moments = (float*)d_ws;             // B*16 floats

    // zero moment accumulators
    int zn = B * 16;
    bp_zero_kernel<<<(zn + 255) / 256, 256, 0, stream>>>(moments, zn);

    // phase 1: segmented moments (4096 waves, coalesced lane-interleaved chunks)
    bp_moments_kernel<<<512, 256, 0, stream>>>(src, tgt, bidx, moments, N);

    // phase 2: per-segment Kabsch (writes R,t into the output tail)
    bp_kabsch_kernel<<<(B + 255) / 256, 256, 0, stream>>>(moments, outR, outT, B);

    // phase 3: apply transform via WMMA 16x16x4 f32 tiles (A cached per segment)
    bp_apply_kernel<<<1024, 256, 0, stream>>>(src, bidx, outR, outT, out_aligned, N);
}